// QuantumCell_89936615178289
// MI455X (gfx1250) — compile-verified
//
#include <hip/hip_runtime.h>

// Problem constants (match reference: B=128, T=2048, D=4, dt=0.02)
#define B_   128
#define T_   2048
#define NCH  128          // number of chunks per batch
#define CL   16           // chunk length: T_/NCH
#define DTC  0.02f

typedef float v2f __attribute__((ext_vector_type(2)));
typedef float v8f __attribute__((ext_vector_type(8)));

// 4x4 complex matmul, fully unrolled so arrays live in registers.
__device__ __forceinline__ void cmm4(const float* ar, const float* ai,
                                     const float* br, const float* bi,
                                     float* cr, float* ci) {
#pragma unroll
  for (int i = 0; i < 4; ++i) {
#pragma unroll
    for (int j = 0; j < 4; ++j) {
      float xr = 0.f, xi = 0.f;
#pragma unroll
      for (int k = 0; k < 4; ++k) {
        const float a_r = ar[i * 4 + k], a_i = ai[i * 4 + k];
        const float b_r = br[k * 4 + j], b_i = bi[k * 4 + j];
        xr = fmaf(a_r, b_r, xr);
        xr = fmaf(-a_i, b_i, xr);
        xi = fmaf(a_r, b_i, xi);
        xi = fmaf(a_i, b_r, xi);
      }
      cr[i * 4 + j] = xr;
      ci[i * 4 + j] = xi;
    }
  }
}

// ---------------------------------------------------------------------------
// K1: per (b, chunk): expm of each step (Taylor/Horner, 8 terms) and local
// prefix products Lp[b,c,l] = U_{cL+l} ... U_{cL}, stored to workspace as
// [b][c][l][ real 4x4 row-major | imag 4x4 row-major ]  (32 floats each).
// ---------------------------------------------------------------------------
__global__ void __launch_bounds__(256)
k1_expm_localprefix(const float* __restrict__ hr_g,
                    const float* __restrict__ hi_g,
                    float* __restrict__ lp) {
  const int tid = blockIdx.x * blockDim.x + threadIdx.x;  // = b*NCH + c
  if (tid >= B_ * NCH) return;
  const int b = tid / NCH;
  const int c = tid % NCH;

  float pr[16], pi[16];
#pragma unroll
  for (int e = 0; e < 16; ++e) {
    pr[e] = ((e & 3) == (e >> 2)) ? 1.f : 0.f;   // P = I
    pi[e] = 0.f;
  }

  const size_t t0 = (size_t)b * T_ + (size_t)c * CL;
  const float* hr = hr_g + t0 * 16;
  const float* hi = hi_g + t0 * 16;
  float* outp = lp + (size_t)tid * CL * 32;

#pragma unroll 1
  for (int l = 0; l < CL; ++l) {
    if (l + 1 < CL) {                       // CDNA5 global_prefetch_b8
      __builtin_prefetch(hr + (size_t)(l + 1) * 16, 0, 3);
      __builtin_prefetch(hi + (size_t)(l + 1) * 16, 0, 3);
    }
    // M = -i*dt*(Hr + i*Hi)  ->  Mr = dt*Hi , Mi = -dt*Hr
    float mr[16], mi[16];
#pragma unroll
    for (int e = 0; e < 16; ++e) {
      const float hrv = hr[(size_t)l * 16 + e];
      const float hiv = hi[(size_t)l * 16 + e];
      mr[e] = DTC * hiv;
      mi[e] = -DTC * hrv;
    }
    // Horner Taylor: S = I + M(I + M/2(I + ... (I + M/8) ...))
    float sr[16], si[16];
#pragma unroll
    for (int e = 0; e < 16; ++e) {
      sr[e] = ((e & 3) == (e >> 2)) ? 1.f : 0.f;
      si[e] = 0.f;
    }
#pragma unroll
    for (int k = 8; k >= 1; --k) {
      float tr[16], ti[16];
      cmm4(mr, mi, sr, si, tr, ti);
      const float inv = 1.0f / (float)k;    // compile-time constant (unrolled)
#pragma unroll
      for (int e = 0; e < 16; ++e) {
        sr[e] = fmaf(tr[e], inv, ((e & 3) == (e >> 2)) ? 1.f : 0.f);
        si[e] = ti[e] * inv;
      }
    }
    // P = E @ P ; store local prefix
    float nr[16], ni[16];
    cmm4(sr, si, pr, pi, nr, ni);
#pragma unroll
    for (int e = 0; e < 16; ++e) {
      pr[e] = nr[e];
      pi[e] = ni[e];
      outp[(size_t)l * 32 + e] = nr[e];
      outp[(size_t)l * 32 + 16 + e] = ni[e];
    }
  }
}

// ---------------------------------------------------------------------------
// K2: per batch b, exclusive scan of chunk totals applied to state0:
// Sb[b,c] = G[b,c-1] ... G[b,0] @ state0[b]   (Sb[b,0] = state0[b]).
// Chunk total G[b,c] = Lp[b,c,CL-1]. Output layout: [b][c][ r16 | i16 ].
// ---------------------------------------------------------------------------
__global__ void __launch_bounds__(128)
k2_carry_scan(const float* __restrict__ s_real,
              const float* __restrict__ s_imag,
              const float* __restrict__ lp,
              float* __restrict__ sb) {
  const int b = blockIdx.x * blockDim.x + threadIdx.x;
  if (b >= B_) return;

  float sr[16], si[16];
#pragma unroll
  for (int e = 0; e < 16; ++e) {
    sr[e] = s_real[(size_t)b * 16 + e];
    si[e] = s_imag[(size_t)b * 16 + e];
  }

#pragma unroll 1
  for (int c = 0; c < NCH; ++c) {
    float* o = sb + ((size_t)b * NCH + c) * 32;
    const float* g = lp + (((size_t)b * NCH + c) * CL + (CL - 1)) * 32;
    if (c + 1 < NCH)
      __builtin_prefetch(lp + (((size_t)b * NCH + c + 1) * CL + (CL - 1)) * 32, 0, 3);
    float gr[16], gi[16];
#pragma unroll
    for (int e = 0; e < 16; ++e) {
      o[e] = sr[e];
      o[16 + e] = si[e];
      gr[e] = g[e];
      gi[e] = g[16 + e];
    }
    float nr[16], ni[16];
    cmm4(gr, gi, sr, si, nr, ni);
#pragma unroll
    for (int e = 0; e < 16; ++e) { sr[e] = nr[e]; si[e] = ni[e]; }
  }
}

// ---------------------------------------------------------------------------
// K3: out[b, c*CL+l] = Lp[b,c,l] @ Sb[b,c], via f32 WMMA.
// Complex 4x4 matmul == 8x8 real rep [[Lr,-Li],[Li,Lr]] @ [Sr;Si].
// Two l-values stacked vertically: A = [R(Lp_l0); R(Lp_l1)] (16x8),
// B = [ [Sr;Si] (8x4) | zeros ] (8x16) -> K=8, just 2x v_wmma_f32_16x16x4_f32.
// C rows 0-7 = result l0, rows 8-15 = result l1 (cols 0-3).
// One wave per (b,c) chunk.
// ---------------------------------------------------------------------------
__global__ void __launch_bounds__(256)
k3_apply_wmma(const float* __restrict__ lp,
              const float* __restrict__ sb,
              float* __restrict__ out) {
  const int wave = blockIdx.x * (blockDim.x >> 5) + (threadIdx.x >> 5);
  const int lane = threadIdx.x & 31;
  const int b = wave / NCH;
  const int c = wave % NCH;

  // A operand (16x8): lane n (low half) feeds row n, K={0,1} (slice0) and
  // K={4,5} (slice1); lane 16+n feeds row n, K={2,3} and K={6,7}.
  const int n   = lane & 15;        // A row / B-C column index
  const int hi  = lane >> 4;        // lane half: K/col sub-pair selector
  const int sel = (n >> 3) & 1;     // 0 -> l0 rows (0-7), 1 -> l1 rows (8-15)
  const int r   = n & 7;            // row within the 8x8 real rep
  const int hlf = r >> 2;           // 0: (Lr,-Li) row half, 1: (Li,Lr)
  const int rp  = r & 3;            // row within 4x4
  const int c0  = 2 * hi;           // column pair base within 4

  // ---- B operand (constant over the chunk): [Sr;Si] in cols 0-3 only.
  // lane n<16: b0=B[0,n], b1=B[1,n] (slice0), b2=B[4,n], b3=B[5,n] (slice1)
  // lane 16+n: b0=B[2,n], b1=B[3,n],          b2=B[6,n], b3=B[7,n]
  const float* sbp = sb + ((size_t)b * NCH + c) * 32;
  const int  j      = n & 3;
  const bool bvalid = (n < 4);
  const int  k0     = 2 * hi;
  const float b0 = bvalid ? sbp[k0 * 4 + j]            : 0.f;  // Sr[k0  ,j]
  const float b1 = bvalid ? sbp[(k0 + 1) * 4 + j]      : 0.f;  // Sr[k0+1,j]
  const float b2 = bvalid ? sbp[16 + k0 * 4 + j]       : 0.f;  // Si[k0  ,j]
  const float b3 = bvalid ? sbp[16 + (k0 + 1) * 4 + j] : 0.f;  // Si[k0+1,j]

  const float* lpbase = lp + ((size_t)b * NCH + c) * CL * 32;

#pragma unroll 1
  for (int g = 0; g < CL / 2; ++g) {
    const int l = 2 * g + sel;                  // which matrix this lane feeds
    const float* m = lpbase + (size_t)l * 32;
    const float prv0 = m[rp * 4 + c0];
    const float prv1 = m[rp * 4 + c0 + 1];
    const float piv0 = m[16 + rp * 4 + c0];
    const float piv1 = m[16 + rp * 4 + c0 + 1];
    // row r of [[Lr,-Li],[Li,Lr]] at local cols {c0,c0+1} and {4+c0,4+c0+1}
    const float a0 = hlf ? piv0 : prv0;         // K slice 0
    const float a1 = hlf ? piv1 : prv1;
    const float a2 = hlf ? prv0 : -piv0;        // K slice 1
    const float a3 = hlf ? prv1 : -piv1;

    v8f acc = {0.f, 0.f, 0.f, 0.f, 0.f, 0.f, 0.f, 0.f};
    v2f av, bv;
    av.x = a0; av.y = a1; bv.x = b0; bv.y = b1;
    acc = __builtin_amdgcn_wmma_f32_16x16x4_f32(false, av, false, bv,
                                                (short)0, acc, false, false);
    av.x = a2; av.y = a3; bv.x = b2; bv.y = b3;
    acc = __builtin_amdgcn_wmma_f32_16x16x4_f32(false, av, false, bv,
                                                (short)0, acc, false, false);

    // Results (cols 0-3): lanes 0-3 hold l0 (rows 0-7): acc[0..3]=Cr rows,
    // acc[4..7]=Ci rows; lanes 16-19 hold l1 (rows 8-15) at col lane-16.
    const bool active = ((lane & 15) < 4);
    if (active) {
      const int tsel = lane >> 4;               // 0 -> l0, 1 -> l1
      const int jj = lane & 3;
      const int t = c * CL + 2 * g + tsel;
      const size_t ro = ((size_t)t * B_ + b) * 16 + jj;          // real plane
      const size_t io = (size_t)T_ * B_ * 16 + ro;               // imag plane
      out[ro + 0]  = acc[0];
      out[ro + 4]  = acc[1];
      out[ro + 8]  = acc[2];
      out[ro + 12] = acc[3];
      out[io + 0]  = acc[4];
      out[io + 4]  = acc[5];
      out[io + 8]  = acc[6];
      out[io + 12] = acc[7];
    }
  }
}

// ---------------------------------------------------------------------------
extern "C" void kernel_launch(void* const* d_in, const int* in_sizes, int n_in,
                              void* d_out, int out_size, void* d_ws,
                              size_t ws_size, hipStream_t stream) {
  (void)in_sizes; (void)n_in; (void)out_size; (void)ws_size;
  const float* h_real = (const float*)d_in[0];
  const float* h_imag = (const float*)d_in[1];
  const float* s_real = (const float*)d_in[2];
  const float* s_imag = (const float*)d_in[3];
  float* out = (float*)d_out;

  // Workspace: Lp = B*T*32 floats (33.5 MB), Sb = B*NCH*32 floats (2 MB).
  float* lp = (float*)d_ws;
  float* sb = lp + (size_t)B_ * T_ * 32;

  k1_expm_localprefix<<<(B_ * NCH + 255) / 256, 256, 0, stream>>>(h_real,
                                                                  h_imag, lp);
  k2_carry_scan<<<1, 128, 0, stream>>>(s_real, s_imag, lp, sb);
  k3_apply_wmma<<<(B_ * NCH) / 8, 256, 0, stream>>>(lp, sb, out);
}